// LocalGaussianRefinement_63050119905588
// MI455X (gfx1250) — compile-verified
//
#include <hip/hip_runtime.h>
#include <hip/hip_bf16.h>

// Problem constants (match reference): B=128, K=17, H=192, W=256, r=2
#define BB   128
#define KKP  17
#define HH   192
#define WW   256
#define RAD  2
#define NEGV (-1e30f)

typedef __attribute__((ext_vector_type(2))) float v2f;
typedef __attribute__((ext_vector_type(8))) float v8f;

// One wave32 handles 16 keypoints. Lane l covers keypoint (l&15); the lane
// half (l>>4) selects which K-pair of the 16x4 f32 WMMA A-matrix it feeds.
// 25 window elements padded to 28 = 7 WMMA steps of K=4.
// Three f32 WMMA accumulators reduce: sum(e), sum(e*x), sum(e*y).
__global__ __launch_bounds__(256) void
LocalGaussianRefinement_kernel(const float* __restrict__ hm,
                               const float* __restrict__ cc,
                               float* __restrict__ out)
{
    const int TOTAL = BB * KKP;                 // 2176 keypoints
    const int lane  = threadIdx.x & 31;
    const int wave  = blockIdx.x * (blockDim.x >> 5) + (threadIdx.x >> 5);
    const int taskBase = wave * 16;
    if (taskBase >= TOTAL) return;              // wave-uniform guard

    const int half = lane >> 4;                 // 0: K=0,1   1: K=2,3
    const int row  = lane & 15;                 // keypoint within group
    const int task = taskBase + row;

    // ---- coarse coords -> rounded, clipped integer center (RNE like jnp.round)
    const float cx = cc[task * 2 + 0];
    const float cy = cc[task * 2 + 1];
    int px = (int)rintf(cx); px = px < 0 ? 0 : (px > WW - 1 ? WW - 1 : px);
    int py = (int)rintf(cy); py = py < 0 ? 0 : (py > HH - 1 ? HH - 1 : py);

    const float* base = hm + (size_t)task * (size_t)(HH * WW);

    // ---- gather this lane's 14 window elements (masked logits + clipped coords)
    float val[7][2], xf[7][2], yf[7][2];
#pragma unroll
    for (int j = 0; j < 7; ++j) {
#pragma unroll
        for (int i = 0; i < 2; ++i) {
            const int k  = 4 * j + 2 * half + i;   // 0..27 (>=25 is padding)
            const int dy = k / 5;
            const int dx = k - dy * 5;
            const int xs = px + dx - RAD;
            const int ys = py + dy - RAD;
            const bool ok = (k < 25) & (xs >= 0) & (xs < WW) & (ys >= 0) & (ys < HH);
            const int xc = xs < 0 ? 0 : (xs > WW - 1 ? WW - 1 : xs);
            const int yc = ys < 0 ? 0 : (ys > HH - 1 ? HH - 1 : ys);
            const float v = base[yc * WW + xc];
            val[j][i] = ok ? v : NEGV;
            xf[j][i]  = (float)xc;
            yf[j][i]  = (float)yc;
        }
    }

    // ---- softmax max: 14 local elements, then merge the two lane-halves
    float m = val[0][0];
#pragma unroll
    for (int j = 0; j < 7; ++j) {
        m = fmaxf(m, val[j][0]);
        m = fmaxf(m, val[j][1]);
    }
    m = fmaxf(m, __shfl_xor(m, 16, 32));        // both halves of keypoint agree

    // ---- 7 steps x 3 accumulating V_WMMA_F32_16X16X4_F32 (B = all-ones)
    v8f accE = {}; v8f accX = {}; v8f accY = {};
    v2f ones; ones[0] = 1.0f; ones[1] = 1.0f;   // all-ones B: layout-invariant

#pragma unroll
    for (int j = 0; j < 7; ++j) {
        const float e0 = expf(val[j][0] - m);   // masked -> exp(-1e30) == 0
        const float e1 = expf(val[j][1] - m);
        v2f aE; aE[0] = e0;             aE[1] = e1;
        v2f aX; aX[0] = e0 * xf[j][0];  aX[1] = e1 * xf[j][1];
        v2f aY; aY[0] = e0 * yf[j][0];  aY[1] = e1 * yf[j][1];
        accE = __builtin_amdgcn_wmma_f32_16x16x4_f32(false, aE, false, ones,
                                                     (short)0, accE, false, false);
        accX = __builtin_amdgcn_wmma_f32_16x16x4_f32(false, aX, false, ones,
                                                     (short)0, accX, false, false);
        accY = __builtin_amdgcn_wmma_f32_16x16x4_f32(false, aY, false, ones,
                                                     (short)0, accY, false, false);
    }

    // ---- extract: VGPR r of D holds keypoint (r + 8*half); every column equal.
    if (row == 0) {
#pragma unroll
        for (int r = 0; r < 8; ++r) {
            const int t = taskBase + half * 8 + r;
            const float inv = 1.0f / accE[r];
            out[t * 2 + 0] = accX[r] * inv;
            out[t * 2 + 1] = accY[r] * inv;
        }
    }
}

extern "C" void kernel_launch(void* const* d_in, const int* in_sizes, int n_in,
                              void* d_out, int out_size, void* d_ws, size_t ws_size,
                              hipStream_t stream)
{
    (void)in_sizes; (void)n_in; (void)out_size; (void)d_ws; (void)ws_size;
    const float* hm = (const float*)d_in[0];   // heatmaps  (B,K,H,W) f32
    const float* cc = (const float*)d_in[1];   // coarse_coords (B,K,2) f32
    float* out = (float*)d_out;                // (B,K,2) f32

    const int totalTasks    = BB * KKP;                    // 2176
    const int wavesNeeded   = (totalTasks + 15) / 16;      // 136
    const int wavesPerBlock = 8;                           // 256 threads, wave32
    const int blocks = (wavesNeeded + wavesPerBlock - 1) / wavesPerBlock; // 17

    hipLaunchKernelGGL(LocalGaussianRefinement_kernel,
                       dim3(blocks), dim3(256), 0, stream, hm, cc, out);
}